// QuadConv_56710748176764
// MI455X (gfx1250) — compile-verified
//
#include <hip/hip_runtime.h>
#include <cstdint>
#include <cstddef>

// ---------------------------------------------------------------------------
// QuadConv for MI455X (gfx1250, wave32).
// Exploits the compact support of the bump kernel (only ~40/2048 neighbors per
// range point contribute) and runs the per-pair sin-MLP through
// V_WMMA_F32_16X16X4_F32 on 16-pair batches. Fully deterministic (no atomics).
// ---------------------------------------------------------------------------

typedef float v2f __attribute__((ext_vector_type(2)));
typedef float v8f __attribute__((ext_vector_type(8)));

#define IN_CH  4
#define OUT_CH 8
#define BATCH  4
#define NBCAP  256     // per-range-point neighbor cap (expected ~40)
#define RHO_CH 1536    // LDS staging chunk for the rho gather

__device__ __forceinline__ float sigmoidf_(float x) {
  return 1.0f / (1.0f + __expf(-x));
}

// D = A(16x4) * B(4x16) + C  (fp32 WMMA, exact arithmetic)
__device__ __forceinline__ v8f wmma4(v2f a, v2f b, v8f c) {
  return __builtin_amdgcn_wmma_f32_16x16x4_f32(
      /*neg_a=*/false, a, /*neg_b=*/false, b,
      /*c_mod=*/(short)0, c, /*reuse_a=*/false, /*reuse_b=*/false);
}

// A-operand loader. Reference reinterprets flat (in,out) weights as (out,in):
// A[o,k] = wflat[o*in_dim + k].  ISA A layout (16x4 f32):
//   lanes 0-15:  VGPR0=A[m,k0+0], VGPR1=A[m,k0+1]   (k0 = 4g)
//   lanes 16-31: VGPR0=A[m,k0+2], VGPR1=A[m,k0+3]
__device__ __forceinline__ v2f load_a(const float* __restrict__ wf,
                                      int in_dim, int row, int k0) {
  v2f a;
  a.x = (k0     < in_dim) ? wf[row * in_dim + k0]     : 0.0f;
  a.y = (k0 + 1 < in_dim) ? wf[row * in_dim + k0 + 1] : 0.0f;
  return a;
}

// --------------------------- rho stage -------------------------------------

// Per-element weight-map MLP: 6 -> 8 -> 8 -> 8 -> 3, sigmoid everywhere.
__global__ void elw_kernel(const float* __restrict__ dp,
                           const int* __restrict__ adj,
                           const float* __restrict__ w0, const float* __restrict__ b0,
                           const float* __restrict__ w1, const float* __restrict__ b1,
                           const float* __restrict__ w2, const float* __restrict__ b2,
                           const float* __restrict__ w3, const float* __restrict__ b3,
                           float* __restrict__ elw, int E) {
  int e = blockIdx.x * blockDim.x + threadIdx.x;
  if (e >= E) return;
  float p[6];
#pragma unroll
  for (int k = 0; k < 3; k++) {
    int a = adj[3 * e + k];
    p[2 * k]     = dp[2 * a];
    p[2 * k + 1] = dp[2 * a + 1];
  }
  float h0[8], h1[8], h2[8];
#pragma unroll
  for (int j = 0; j < 8; j++) {
    float s = b0[j];
#pragma unroll
    for (int k = 0; k < 6; k++) s += p[k] * w0[k * 8 + j];
    h0[j] = sigmoidf_(s);
  }
#pragma unroll
  for (int j = 0; j < 8; j++) {
    float s = b1[j];
#pragma unroll
    for (int k = 0; k < 8; k++) s += h0[k] * w1[k * 8 + j];
    h1[j] = sigmoidf_(s);
  }
#pragma unroll
  for (int j = 0; j < 8; j++) {
    float s = b2[j];
#pragma unroll
    for (int k = 0; k < 8; k++) s += h1[k] * w2[k * 8 + j];
    h2[j] = sigmoidf_(s);
  }
#pragma unroll
  for (int j = 0; j < 3; j++) {
    float s = b3[j];
#pragma unroll
    for (int k = 0; k < 8; k++) s += h2[k] * w3[k * 3 + j];
    elw[3 * e + j] = sigmoidf_(s);
  }
}

// Deterministic gather (replaces scatter-add): rho[i] = sum over matching slots.
__global__ void rho_kernel(const int* __restrict__ adj,
                           const float* __restrict__ elw,
                           float* __restrict__ rho, int N, int E3) {
  __shared__ int   sadj[RHO_CH];
  __shared__ float selw[RHO_CH];
  int i = blockIdx.x * blockDim.x + threadIdx.x;
  float s = 0.0f;
  for (int base = 0; base < E3; base += RHO_CH) {
    int cnt = E3 - base;
    if (cnt > RHO_CH) cnt = RHO_CH;
    for (int t = threadIdx.x; t < cnt; t += blockDim.x) {
      sadj[t] = adj[base + t];
      selw[t] = elw[base + t];
    }
    __syncthreads();
    for (int t = 0; t < cnt; t++) s += (sadj[t] == i) ? selw[t] : 0.0f;
    __syncthreads();
  }
  if (i < N) rho[i] = s;
}

// --------------------------- main stage ------------------------------------
// One wave32 workgroup per range point j.
__global__ __launch_bounds__(32)
void quadconv_kernel(const float* __restrict__ feat,   // [B,N,4]
                     const float* __restrict__ dp,     // [N,2]
                     const float* __restrict__ rp,     // [M,2]
                     const float* __restrict__ w0, const float* __restrict__ w1,
                     const float* __restrict__ w2, const float* __restrict__ w3,
                     const float* __restrict__ w4, const float* __restrict__ w5,
                     const float* __restrict__ rho,
                     float* __restrict__ out,          // [B,M,8]
                     int N, int M) {
  __shared__ float qdx[NBCAP], qdy[NBCAP];
  __shared__ float qcf[NBCAP][BATCH * IN_CH];  // rho_i * bump * f[b,i,c]
  __shared__ float act[16][16];                // activations, [feature][pair]
  __shared__ float k5[32][16];                 // final layer output

  const int j    = blockIdx.x;
  const int lane = threadIdx.x;
  const int m    = lane & 15;
  const int hi   = lane >> 4;
  const int k0h  = 2 * hi;

  const float yx = rp[2 * j], yy = rp[2 * j + 1];
  const float tt = (float)N * 0.0625f;
  const float alpha = tt * tt;                 // (N/16)^2

  // ---- Phase 1: ballot-compact in-support neighbors into the LDS queue ----
  int qn = 0;
  for (int base_i = 0; base_i < N; base_i += 32) {
    int i = base_i + lane;
    float dx = 0.0f, dy = 0.0f, denom = 1.0f;
    bool inside = false;
    if (i < N) {
      dx = yx - dp[2 * i];
      dy = yy - dp[2 * i + 1];
      float r2 = dx * dx + dy * dy;
      inside = (alpha * r2 < 1.0f);
      denom  = 1.0f - alpha * r2;
    }
    unsigned msk = (unsigned)__ballot(inside);
    if (inside) {
      int slot = qn + __popc(msk & ((1u << lane) - 1u));
      if (slot < NBCAP) {
        qdx[slot] = dx;
        qdy[slot] = dy;
        float w = rho[i] * __expf(-1.0f / denom);   // bump folded into coeff
#pragma unroll
        for (int b = 0; b < BATCH; b++) {
          const float* fp = feat + ((size_t)b * N + i) * IN_CH;
#pragma unroll
          for (int c = 0; c < IN_CH; c++)
            qcf[slot][b * IN_CH + c] = w * fp[c];
        }
      }
    }
    qn += __popc(msk);
    if (qn > NBCAP) qn = NBCAP;
  }
  __syncthreads();

  // ---- Preload all WMMA A-operands (weights) into registers --------------
  v2f A0 = load_a(w0, 2, m, k0h);              // 2->16 layer, K padded to 4
  v2f A14[4][4];
  const float* wl[4] = {w1, w2, w3, w4};
#pragma unroll
  for (int l = 0; l < 4; l++)
#pragma unroll
    for (int g = 0; g < 4; g++)
      A14[l][g] = load_a(wl[l], 16, m, 4 * g + k0h);
  v2f A5[2][4];
#pragma unroll
  for (int h = 0; h < 2; h++)
#pragma unroll
    for (int g = 0; g < 4; g++)
      A5[h][g] = load_a(w5, 16, 16 * h + m, 4 * g + k0h);

  // lane -> (batch, out-channel); exactly 32 combos
  const int ob = lane >> 3;
  const int oo = lane & 7;
  float accO = 0.0f;

  // ---- Phase 2: 16-pair batches through the WMMA sin-MLP -----------------
  for (int base = 0; base < qn; base += 16) {
    const int pv = (qn - base < 16) ? (qn - base) : 16;

    // Layer 0: X1 = sin(A0(16x2) * diff(2x16)), K zero-padded to 4
    v8f d = {0.f, 0.f, 0.f, 0.f, 0.f, 0.f, 0.f, 0.f};
    {
      v2f b;
      bool valid = (hi == 0) && (base + m < qn);
      b.x = valid ? qdx[base + m] : 0.0f;   // B rows {dx, dy, 0, 0}
      b.y = valid ? qdy[base + m] : 0.0f;
      d = wmma4(A0, b, d);
    }
#pragma unroll
    for (int r = 0; r < 8; r++) act[r + 8 * hi][m] = __sinf(d[r]);
    __syncthreads();

    // Layers 1..4: 16x16, 4 chained K=4 WMMAs each
#pragma unroll
    for (int l = 0; l < 4; l++) {
      v8f t = {0.f, 0.f, 0.f, 0.f, 0.f, 0.f, 0.f, 0.f};
#pragma unroll
      for (int g = 0; g < 4; g++) {
        int k0 = 4 * g + k0h;
        v2f b;
        b.x = act[k0][m];
        b.y = act[k0 + 1][m];
        t = wmma4(A14[l][g], b, t);
      }
      __syncthreads();
#pragma unroll
      for (int r = 0; r < 8; r++) act[r + 8 * hi][m] = __sinf(t[r]);
      __syncthreads();
    }

    // Layer 5: 16 -> 32 (two 16-row halves)
    v8f t5[2];
#pragma unroll
    for (int h = 0; h < 2; h++) {
      v8f t = {0.f, 0.f, 0.f, 0.f, 0.f, 0.f, 0.f, 0.f};
#pragma unroll
      for (int g = 0; g < 4; g++) {
        int k0 = 4 * g + k0h;
        v2f b;
        b.x = act[k0][m];
        b.y = act[k0 + 1][m];
        t = wmma4(A5[h][g], b, t);
      }
      t5[h] = t;
    }
    __syncthreads();
#pragma unroll
    for (int r = 0; r < 8; r++) {
      k5[r + 8 * hi][m]      = __sinf(t5[0][r]);
      k5[16 + r + 8 * hi][m] = __sinf(t5[1][r]);
    }
    __syncthreads();

    // Contract: acc[b,o] += sum_{p,c} qcf[p][b,c] * k5[c*8+o][p]
    for (int p = 0; p < pv; p++) {
      const float* cf = &qcf[base + p][ob * IN_CH];
      float s = 0.0f;
#pragma unroll
      for (int c = 0; c < IN_CH; c++) s += cf[c] * k5[c * OUT_CH + oo][p];
      accO += s;
    }
    __syncthreads();
  }

  // One deterministic store per lane: out[b, j, o]
  out[((size_t)ob * M + j) * OUT_CH + oo] = accO;
}

// --------------------------- launcher --------------------------------------

extern "C" void kernel_launch(void* const* d_in, const int* in_sizes, int n_in,
                              void* d_out, int out_size, void* d_ws, size_t ws_size,
                              hipStream_t stream) {
  const float* feat = (const float*)d_in[0];
  const float* dp   = (const float*)d_in[1];
  const float* rp   = (const float*)d_in[2];
  const float* w0   = (const float*)d_in[3];
  const float* w1   = (const float*)d_in[4];
  const float* w2   = (const float*)d_in[5];
  const float* w3   = (const float*)d_in[6];
  const float* w4   = (const float*)d_in[7];
  const float* w5   = (const float*)d_in[8];
  const float* mw0  = (const float*)d_in[9];
  const float* mb0  = (const float*)d_in[10];
  const float* mw1  = (const float*)d_in[11];
  const float* mb1  = (const float*)d_in[12];
  const float* mw2  = (const float*)d_in[13];
  const float* mb2  = (const float*)d_in[14];
  const float* mw3  = (const float*)d_in[15];
  const float* mb3  = (const float*)d_in[16];
  const int*   adj  = (const int*)d_in[17];

  const int N = in_sizes[1] / 2;    // domain points
  const int M = in_sizes[2] / 2;    // range points
  const int E = in_sizes[17] / 3;   // elements

  float* rho = (float*)d_ws;        // [N]
  float* elw = rho + N;             // [E*3]

  elw_kernel<<<(E + 255) / 256, 256, 0, stream>>>(
      dp, adj, mw0, mb0, mw1, mb1, mw2, mb2, mw3, mb3, elw, E);
  rho_kernel<<<(N + 255) / 256, 256, 0, stream>>>(adj, elw, rho, N, E * 3);
  quadconv_kernel<<<M, 32, 0, stream>>>(
      feat, dp, rp, w0, w1, w2, w3, w4, w5, rho, (float*)d_out, N, M);
}